// RewardTripletLoss_68745246540430
// MI455X (gfx1250) — compile-verified
//
#include <hip/hip_runtime.h>
#include <hip/hip_bf16.h>
#include <math.h>

// Problem sizes fixed by setup_inputs(): n = m = 4096, d = 128, bins = 1600.
#define N 4096
#define D 128
#define NBINS 1601           // NUM_BINS + 1
#define INV_DELTA 400.0f     // 1 / (4/1600)
#define MARGIN 0.1f

typedef __attribute__((ext_vector_type(2))) float v2f;
typedef __attribute__((ext_vector_type(8))) float v8f;

// ---------------------------------------------------------------------------
// Kernel 1/3: S = A * B^T  (A:[N,D], B:[N,D] row-major, S:[N,N])
// f32 WMMA 16x16x4. 256 threads = 8 waves; each wave owns one 16x16 tile.
// Grid: (N/16)*(N/16)/8 = 8192 blocks.
// ---------------------------------------------------------------------------
__global__ __launch_bounds__(256) void wmma_gemm_abt(const float* __restrict__ A,
                                                     const float* __restrict__ B,
                                                     float* __restrict__ S) {
    const int lane = threadIdx.x & 31;
    const int wave = threadIdx.x >> 5;
    const int tiles_n = N / 16;                       // 256
    const int tile = blockIdx.x * 8 + wave;           // 0 .. 65535
    const int tm = tile / tiles_n;
    const int tn = tile - tm * tiles_n;

    // A fragment: lane -> row m = tm*16 + (lane&15); K pair = 2*(lane>>4)
    // B fragment: lane -> col n = tn*16 + (lane&15); same K pair (B = X^T)
    const int kb = (lane >> 4) << 1;                  // 0 or 2
    const float* arow = A + (size_t)(tm * 16 + (lane & 15)) * D + kb;
    const float* brow = B + (size_t)(tn * 16 + (lane & 15)) * D + kb;

    v8f c = {};
#pragma unroll 4
    for (int k = 0; k < D; k += 4) {
        v2f a = *(const v2f*)(arow + k);
        v2f b = *(const v2f*)(brow + k);
        // 8 args: (neg_a, A, neg_b, B, c_mod, C, reuse_a, reuse_b)
        c = __builtin_amdgcn_wmma_f32_16x16x4_f32(false, a, false, b,
                                                  (short)0, c, false, false);
    }

    // D layout: VGPR r, lane l -> M = r + 8*(l>>4), N = l&15
    const int mrow = tm * 16 + ((lane >> 4) << 3);
    const int ncol = tn * 16 + (lane & 15);
    float* out = S + (size_t)mrow * N + ncol;
#pragma unroll
    for (int r = 0; r < 8; ++r)
        out[(size_t)r * N] = c[r];
}

// ---------------------------------------------------------------------------
// Kernel 2: per-row triplet-style loss. One 256-thread block per row.
// Row cached in 16 registers/thread; pos/neg kept as 16-bit masks.
// ---------------------------------------------------------------------------
__global__ __launch_bounds__(256) void row_loss_kernel(const float* __restrict__ S,
                                                       const int* __restrict__ tcol,
                                                       const int* __restrict__ trow,
                                                       float* __restrict__ loss) {
    const int i = blockIdx.x;
    const int tid = threadIdx.x;
    const int ti = tcol[i];
    const float* srow = S + (size_t)i * N;

    float sv[16];
    unsigned posm = 0u, negm = 0u;
    float pmax = -INFINITY, nmax = -INFINITY;
#pragma unroll
    for (int r = 0; r < 16; ++r) {
        const int j = tid + (r << 8);
        const float s = srow[j];
        sv[r] = s;
        const bool same = (trow[j] == ti);
        if (same) {
            if (j != i) { posm |= (1u << r); pmax = fmaxf(pmax, s); }
        } else {
            negm |= (1u << r); nmax = fmaxf(nmax, s);
        }
    }

    __shared__ float red[256];
    // max-reduce pmax
    red[tid] = pmax; __syncthreads();
    for (int off = 128; off > 0; off >>= 1) {
        if (tid < off) red[tid] = fmaxf(red[tid], red[tid + off]);
        __syncthreads();
    }
    pmax = red[0]; __syncthreads();
    // max-reduce nmax
    red[tid] = nmax; __syncthreads();
    for (int off = 128; off > 0; off >>= 1) {
        if (tid < off) red[tid] = fmaxf(red[tid], red[tid + off]);
        __syncthreads();
    }
    nmax = red[0]; __syncthreads();

    const bool has_pos = (pmax != -INFINITY);
    const float thr_pos = nmax + MARGIN;              // pos kept if s <  thr_pos
    const float thr_neg = fmaxf(0.6f, pmax) - MARGIN; // neg kept if s >  thr_neg

    float acc = 0.0f;
#pragma unroll
    for (int r = 0; r < 16; ++r) {
        const float s = sv[r];
        if (((posm >> r) & 1u) && (s < thr_pos)) acc += 1.0f - s;
        if (((negm >> r) & 1u) && (s > thr_neg)) acc += s;
    }
    red[tid] = acc; __syncthreads();
    for (int off = 128; off > 0; off >>= 1) {
        if (tid < off) red[tid] += red[tid + off];
        __syncthreads();
    }
    if (tid == 0) loss[i] = has_pos ? red[0] : 0.0f;
}

// ---------------------------------------------------------------------------
// Kernel 4: FastAP reward per row. One 256-thread block per row.
// Two 1601-bin LDS histograms (pos, all), triangular 2-bin scatter via
// ds_add_f32 atomics, chunked block scan (7 bins/thread), AP sum.
// ---------------------------------------------------------------------------
__device__ __forceinline__ void block_scan_1601(float* h, float* part, int tid) {
    const int start = tid * 7;
    float run = 0.0f;
    if (start < NBINS) {
        const int end = min(start + 7, NBINS);
        for (int b = start; b < end; ++b) { run += h[b]; h[b] = run; }
    }
    part[tid] = run;
    __syncthreads();
    if (tid == 0) {
        float a = 0.0f;
        for (int t = 0; t < 256; ++t) { const float v = part[t]; part[t] = a; a += v; }
    }
    __syncthreads();
    if (start < NBINS) {
        const float off = part[tid];
        const int end = min(start + 7, NBINS);
        for (int b = start; b < end; ++b) h[b] += off;
    }
    __syncthreads();
}

__global__ __launch_bounds__(256) void fastap_kernel(const float* __restrict__ G,
                                                     const int* __restrict__ lab,
                                                     float* __restrict__ rewards) {
    const int i = blockIdx.x;
    const int tid = threadIdx.x;
    __shared__ float hp[NBINS];
    __shared__ float ha[NBINS];
    __shared__ float part[256];

    for (int b = tid; b < NBINS; b += 256) { hp[b] = 0.0f; ha[b] = 0.0f; }
    __syncthreads();

    const int li = lab[i];
    const float* grow = G + (size_t)i * N;
    for (int j = tid; j < N; j += 256) {
        const bool same = (lab[j] == li);
        const bool ip = same && (j != i);
        const bool in_ = !same;
        if (!ip && !in_) continue;                    // diagonal-with-same-label
        const float g = grow[j];
        const float d2 = fminf(fmaxf(2.0f - 2.0f * g, 0.0f), 4.0f);
        const float t = d2 * INV_DELTA;
        const float fl = floorf(t);
        int i0 = (int)fl;
        i0 = max(0, min(i0, NBINS - 1));
        const int i1 = min(i0 + 1, NBINS - 1);
        const float frac = t - fl;
        const float w0 = 1.0f - frac;
        atomicAdd(&ha[i0], w0);
        atomicAdd(&ha[i1], frac);
        if (ip) {
            atomicAdd(&hp[i0], w0);
            atomicAdd(&hp[i1], frac);
        }
    }
    __syncthreads();

    block_scan_1601(hp, part, tid);   // hp -> inclusive cumsum H_pos
    block_scan_1601(ha, part, tid);   // ha -> inclusive cumsum H_all

    float ap = 0.0f;
    for (int b = tid; b < NBINS; b += 256) {
        const float Hp = hp[b];
        const float Ha = ha[b];
        const float h = Hp - (b ? hp[b - 1] : 0.0f);
        if (Ha > 0.0f) ap += h * Hp / Ha;
    }
    part[tid] = ap; __syncthreads();
    for (int off = 128; off > 0; off >>= 1) {
        if (tid < off) part[tid] += part[tid + off];
        __syncthreads();
    }
    if (tid == 0) {
        const float npos = hp[NBINS - 1];             // sum of soft pos weights = count
        rewards[i] = (npos > 0.0f) ? (part[0] / fmaxf(npos, 1.0f)) : 0.0f;
    }
}

// ---------------------------------------------------------------------------
// Kernel 5: final scalar: sum_i loss[i] * (1 - rewards[i]) / N
// ---------------------------------------------------------------------------
__global__ __launch_bounds__(256) void final_reduce_kernel(const float* __restrict__ loss,
                                                           const float* __restrict__ rewards,
                                                           float* __restrict__ out) {
    __shared__ float red[256];
    const int tid = threadIdx.x;
    float acc = 0.0f;
    for (int i = tid; i < N; i += 256)
        acc += loss[i] * (1.0f - rewards[i]);
    red[tid] = acc; __syncthreads();
    for (int off = 128; off > 0; off >>= 1) {
        if (tid < off) red[tid] += red[tid + off];
        __syncthreads();
    }
    if (tid == 0) out[0] = red[0] / (float)N;
}

// ---------------------------------------------------------------------------
extern "C" void kernel_launch(void* const* d_in, const int* in_sizes, int n_in,
                              void* d_out, int out_size, void* d_ws, size_t ws_size,
                              hipStream_t stream) {
    const float* x_col = (const float*)d_in[0];   // [N, D]
    const int*   t_col = (const int*)d_in[1];     // [N]
    const float* x_row = (const float*)d_in[2];   // [N, D]
    const int*   t_row = (const int*)d_in[3];     // [N]
    const int*   r_lab = (const int*)d_in[4];     // [N]
    float* out = (float*)d_out;

    // Workspace: S [N*N] f32 (64 MB, reused for sim then Gram) + loss + rewards.
    float* S       = (float*)d_ws;
    float* loss    = (float*)((char*)d_ws + (size_t)N * N * sizeof(float));
    float* rewards = loss + N;

    const int gemm_blocks = (N / 16) * (N / 16) / 8;  // 8192

    // 1) sim = x_col @ x_row^T
    wmma_gemm_abt<<<gemm_blocks, 256, 0, stream>>>(x_col, x_row, S);
    // 2) per-row masked maxima + masked sums -> loss[i]
    row_loss_kernel<<<N, 256, 0, stream>>>(S, t_col, t_row, loss);
    // 3) Gram = x_col @ x_col^T (overwrites S; stream-ordered, so safe)
    wmma_gemm_abt<<<gemm_blocks, 256, 0, stream>>>(x_col, x_col, S);
    // 4) FastAP rewards from Gram rows
    fastap_kernel<<<N, 256, 0, stream>>>(S, r_lab, rewards);
    // 5) scalar output
    final_reduce_kernel<<<1, 256, 0, stream>>>(loss, rewards, out);
}